// UniformSubseqSampler_35270271434844
// MI455X (gfx1250) — compile-verified
//
#include <hip/hip_runtime.h>
#include <hip/hip_bf16.h>
#include <stdint.h>
#include <stddef.h>

// ---------------------------------------------------------------------------
// Uniform subsequence sampler (state machine), MI455X / gfx1250.
//   - per-state legal-token set as 128-bit mask + popcount (prep kernel)
//   - 8-bit transition table (states are 0..63)
//   - tables staged to LDS via CDNA5 async load-to-LDS (ASYNCcnt path)
//   - one row per lane, wave32, sequential 64-step scan
// ---------------------------------------------------------------------------

#define N_STATES   64
#define VOCAB      128
#define MAX_STEPS  64

// Workspace layout (built by uss_prep, consumed by uss_sample)
#define WS_MASKS_OFF   0          // uint4[64]            : 1024 B
#define WS_COUNTS_OFF  1024       // uint32[64]           :  256 B
#define WS_TRANS_OFF   1280       // uint8[64*128]        : 8192 B
#define WS_TOTAL       9472       // == 592 * 16 bytes

#if defined(__gfx1250__) && \
    __has_builtin(__builtin_amdgcn_global_load_async_to_lds_b128) && \
    __has_builtin(__builtin_amdgcn_s_wait_asynccnt)
#define USE_ASYNC_LDS 1
#else
#define USE_ASYNC_LDS 0
#endif

// 16-byte int vector type expected by the async-to-LDS builtin on this toolchain
typedef __attribute__((__vector_size__(4 * sizeof(int)))) int v4i_t;

// --- prep: binary mask rows -> bitmask + popcount; int32 trans -> u8 --------
__global__ void uss_prep_kernel(const float* __restrict__ dm,
                                const int* __restrict__ st,
                                unsigned char* __restrict__ ws) {
    int s = threadIdx.x;
    if (s >= N_STATES) return;
    unsigned int w[4] = {0u, 0u, 0u, 0u};
    unsigned int c = 0u;
    unsigned char* trans = ws + WS_TRANS_OFF;
    for (int v = 0; v < VOCAB; ++v) {
        float f = dm[s * VOCAB + v];
        unsigned int b = (f > 0.5f) ? 1u : 0u;
        w[v >> 5] |= b << (v & 31);
        c += b;
        trans[s * VOCAB + v] = (unsigned char)st[s * VOCAB + v];
    }
    uint4* masks = (uint4*)(ws + WS_MASKS_OFF);
    unsigned int* counts = (unsigned int*)(ws + WS_COUNTS_OFF);
    masks[s] = make_uint4(w[0], w[1], w[2], w[3]);
    counts[s] = c;
}

__device__ __forceinline__ unsigned int splitmix32(unsigned long long z) {
    z ^= z >> 30; z *= 0xBF58476D1CE4E5B9ull;
    z ^= z >> 27; z *= 0x94D049BB133111EBull;
    z ^= z >> 31;
    return (unsigned int)z;
}

// --- main sampler -----------------------------------------------------------
__global__ void __launch_bounds__(256)
uss_sample_kernel(const unsigned char* __restrict__ ws,
                  int* __restrict__ out_tok,
                  int* __restrict__ out_fin,
                  int num) {
    __shared__ unsigned char s_tab[WS_TOTAL];

#if USE_ASYNC_LDS
    // gfx1250 async global->LDS DMA (tracked on ASYNCcnt)
    for (int i = threadIdx.x; i < WS_TOTAL / 16; i += blockDim.x) {
        __builtin_amdgcn_global_load_async_to_lds_b128(
            (v4i_t*)(ws + i * 16), (v4i_t*)(s_tab + i * 16), 0, 0);
    }
    __builtin_amdgcn_s_wait_asynccnt(0);
    __syncthreads();
#else
    for (int i = threadIdx.x * 16; i < WS_TOTAL; i += blockDim.x * 16) {
        *(uint4*)(s_tab + i) = *(const uint4*)(ws + i);
    }
    __syncthreads();
#endif

    const uint4*         s_masks  = (const uint4*)(s_tab + WS_MASKS_OFF);
    const unsigned int*  s_counts = (const unsigned int*)(s_tab + WS_COUNTS_OFF);
    const unsigned char* s_trans  = s_tab + WS_TRANS_OFF;

    for (int r = blockIdx.x * blockDim.x + threadIdx.x; r < num;
         r += gridDim.x * blockDim.x) {
        unsigned int cur = 3u;   // NON_EMPTY_EXPR
        unsigned int fin = 3u;
        bool active = true;

        unsigned int packed[MAX_STEPS / 4];
        #pragma unroll
        for (int i = 0; i < MAX_STEPS / 4; ++i) packed[i] = 0u;

        #pragma unroll 1
        for (int t = 0; t < MAX_STEPS; ++t) {
            int tokOut = -1;
            if (active) {
                unsigned long long z =
                    0x9E3779B97F4A7C15ull * (unsigned long long)(r + 1) +
                    0xD1B54A32D192ED03ull * (unsigned long long)(t + 1);
                unsigned int rnd  = splitmix32(z);
                uint4        m    = s_masks[cur];
                unsigned int c    = s_counts[cur];
                unsigned int pick = rnd % c;            // uniform over set bits

                unsigned int word = m.x, base = 0u;
                unsigned int pc = (unsigned int)__popc(word);
                if (pick >= pc) {
                    pick -= pc; word = m.y; base = 32u; pc = (unsigned int)__popc(word);
                    if (pick >= pc) {
                        pick -= pc; word = m.z; base = 64u; pc = (unsigned int)__popc(word);
                        if (pick >= pc) { pick -= pc; word = m.w; base = 96u; }
                    }
                }
                #pragma unroll 1
                while (pick--) word &= word - 1u;       // drop `pick` lowest set bits
                unsigned int tok = base + (unsigned int)__ffs(word) - 1u;

                tokOut = (int)tok;
                unsigned int nxt = (unsigned int)s_trans[cur * VOCAB + tok];
                if (nxt <= 1u) { fin = nxt; active = false; }   // START/HALT
                else           { cur = nxt; }
            }
            packed[t >> 2] |= ((unsigned int)(tokOut & 0xFF)) << ((t & 3) * 8);
        }

        // unpack int8 tokens -> int32, 16B vector stores (row owns 256 B)
        #pragma unroll
        for (int i = 0; i < MAX_STEPS / 4; ++i) {
            unsigned int p = packed[i];
            int4 v;
            v.x = (int)(signed char)( p        & 0xFF);
            v.y = (int)(signed char)((p >> 8)  & 0xFF);
            v.z = (int)(signed char)((p >> 16) & 0xFF);
            v.w = (int)(signed char)((p >> 24) & 0xFF);
            *(int4*)(out_tok + (size_t)r * MAX_STEPS + i * 4) = v;
        }
        out_fin[r] = (int)fin;
    }
}

extern "C" void kernel_launch(void* const* d_in, const int* in_sizes, int n_in,
                              void* d_out, int out_size, void* d_ws, size_t ws_size,
                              hipStream_t stream) {
    const float* dm = (const float*)d_in[0];   // decision_mask [64,128] f32
    const int*   st = (const int*)d_in[1];     // state_trans   [64,128] i32
    (void)in_sizes; (void)n_in; (void)ws_size;

    // out = tokens[num,64] ++ final_states[num]  =>  num = out_size / 65
    int num = out_size / (MAX_STEPS + 1);

    unsigned char* ws      = (unsigned char*)d_ws;
    int*           out_tok = (int*)d_out;
    int*           out_fin = out_tok + (size_t)num * MAX_STEPS;

    uss_prep_kernel<<<1, 64, 0, stream>>>(dm, st, ws);

    int threads = 256;                       // 8 wave32 per block
    int blocks  = (num + threads - 1) / threads;
    if (blocks < 1) blocks = 1;
    if (blocks > 4096) blocks = 4096;        // grid-stride covers the rest
    uss_sample_kernel<<<blocks, threads, 0, stream>>>(ws, out_tok, out_fin, num);
}